// GCN_SUB_2774548873595
// MI455X (gfx1250) — compile-verified
//
#include <hip/hip_runtime.h>
#include <hip/hip_bf16.h>

typedef float v2f __attribute__((ext_vector_type(2)));
typedef float v8f __attribute__((ext_vector_type(8)));

#define NNODES 100000
#define NEDGES 1600000

// ---------------------------------------------------------------------------
// Tiled fp32 GEMM + bias using V_WMMA_F32_16X16X4_F32.
// P[N, dout] = (relu_in ? relu(X) : X)[N, din] @ Wm[din, doutp](padded) + bias
// Block = 128 threads = 4 waves; each wave owns one 16-wide column tile.
// A-tile (16 x din) staged in LDS (row pad +4 -> conflict-free banks).
// Wm/bias are pre-padded to doutp (multiple of 16) so the inner loop has NO
// divergence: EXEC stays all-1s, no exec save/restore around the WMMA.
// ---------------------------------------------------------------------------
__global__ __launch_bounds__(128) void gcn_gemm_wmma(
    const float* __restrict__ X, const float* __restrict__ Wm,
    const float* __restrict__ bias, float* __restrict__ P,
    int N, int din, int dout, int doutp, int relu_in)
{
    __shared__ float lds[16 * (128 + 4)];   // max din = 128, +4 pad per row
    const int ldsrow = din + 4;
    const int row0 = blockIdx.x * 16;
    const int t = threadIdx.x;

    // ---- cooperative stage of A tile (float4, relu fused on load) ----
    const int nvec = (16 * din) >> 2;        // din is a multiple of 4
    const int vper = din >> 2;
    for (int idx = t; idx < nvec; idx += 128) {
        const int r  = idx / vper;
        const int c4 = (idx - r * vper) << 2;
        const int row = row0 + r;
        float4 v = make_float4(0.f, 0.f, 0.f, 0.f);
        if (row < N) {
            v = *(const float4*)(X + (size_t)row * din + c4);
            if (relu_in) {
                v.x = fmaxf(v.x, 0.f); v.y = fmaxf(v.y, 0.f);
                v.z = fmaxf(v.z, 0.f); v.w = fmaxf(v.w, 0.f);
            }
        }
        *(float4*)(&lds[r * ldsrow + c4]) = v;
    }
    __syncthreads();

    const int wave = t >> 5;
    const int lane = t & 31;
    const int colbase = wave * 16;
    if (colbase >= doutp) return;            // wave-uniform: EXEC stays all-1s

    const int half = lane >> 4;              // 0: lanes 0-15, 1: lanes 16-31
    const int nrow = lane & 15;              // A: M index ; B/C/D: N index
    const int col  = colbase + nrow;         // always < doutp (padded)

    // running pointers -> no per-iteration 64-bit address rebuilds
    const float* wk = Wm + (size_t)(2 * half) * doutp + col;
    int aoff = nrow * ldsrow + 2 * half;     // lane-half K split per ISA layout

    v8f acc = {};
    for (int k = 0; k < din; k += 4) {
        v2f a, b;
        a.x = lds[aoff];
        a.y = lds[aoff + 1];
        b.x = wk[0];
        b.y = wk[doutp];
        acc = __builtin_amdgcn_wmma_f32_16x16x4_f32(
                  false, a, false, b, (short)0, acc, false, false);
        aoff += 4;
        wk   += (size_t)4 * doutp;
    }

    const float bv = bias[col];              // bias padded to doutp
    #pragma unroll
    for (int r = 0; r < 8; ++r) {
        const int m   = r + 8 * half;        // C/D layout: half 1 holds M=8..15
        const int row = row0 + m;
        if (row < N && col < dout)
            P[(size_t)row * dout + col] = acc[r] + bv;
    }
}

// ---------------------------------------------------------------------------
// Pad W4 (64x10 -> 64x16) and b4 (10 -> 16) with zeros.
// ---------------------------------------------------------------------------
__global__ void gcn_pad_w4(const float* __restrict__ W4, const float* __restrict__ b4,
                           float* __restrict__ Wp, float* __restrict__ bp)
{
    const int i = blockIdx.x * blockDim.x + threadIdx.x;
    if (i < 64 * 16) {
        const int r = i >> 4, c = i & 15;
        Wp[i] = (c < 10) ? W4[r * 10 + c] : 0.f;
    } else if (i < 64 * 16 + 16) {
        const int c = i - 64 * 16;
        bp[c] = (c < 10) ? b4[c] : 0.f;
    }
}

// ---------------------------------------------------------------------------
// Zero-fill
// ---------------------------------------------------------------------------
__global__ void gcn_zero(float* __restrict__ p, int n)
{
    const int i = blockIdx.x * blockDim.x + threadIdx.x;
    if (i < n) p[i] = 0.f;
}

// ---------------------------------------------------------------------------
// Edge aggregation: out[dst[e]] += P[src[e]] * w[e]
// One wave per edge; each lane handles a float2 column pair.
// Native f32 atomics (relaxed, agent scope) -> global_atomic_add_f32.
// Both P and out are L2-resident (25.6 MB each vs 192 MB L2).
// ---------------------------------------------------------------------------
__global__ __launch_bounds__(256) void gcn_aggregate(
    const float* __restrict__ P, const int* __restrict__ src,
    const int* __restrict__ dst, const float* __restrict__ w,
    float* __restrict__ out, int E, int dout)
{
    const int gid = blockIdx.x * blockDim.x + threadIdx.x;
    const int e = gid >> 5;
    if (e >= E) return;
    const int lane = gid & 31;

    // stream-ahead prefetch of the linearly-read edge arrays
    if (lane == 0) {
        const int ep = e + 4096;
        if (ep < E) {
            __builtin_prefetch(src + ep, 0, 0);
            __builtin_prefetch(dst + ep, 0, 0);
            __builtin_prefetch(w   + ep, 0, 0);
        }
    }

    const int c = lane << 1;
    if (c >= dout) return;                   // dout is even (64 or 10)

    const int s = src[e];
    const int d = dst[e];
    const float ww = w[e];

    const float2 h = *(const float2*)(P + (size_t)s * dout + c);
    float* o = out + (size_t)d * dout + c;
    __hip_atomic_fetch_add(o,     h.x * ww, __ATOMIC_RELAXED, __HIP_MEMORY_SCOPE_AGENT);
    __hip_atomic_fetch_add(o + 1, h.y * ww, __ATOMIC_RELAXED, __HIP_MEMORY_SCOPE_AGENT);
}

// ---------------------------------------------------------------------------
extern "C" void kernel_launch(void* const* d_in, const int* in_sizes, int n_in,
                              void* d_out, int out_size, void* d_ws, size_t ws_size,
                              hipStream_t stream)
{
    (void)in_sizes; (void)n_in; (void)out_size; (void)ws_size;
    const int N = NNODES, E = NEDGES;

    const float* x    = (const float*)d_in[0];
    const int*   esrc = (const int*)  d_in[1];
    const int*   edst = (const int*)  d_in[2];
    const float* ew   = (const float*)d_in[3];
    const float* Ws[5]; const float* bs[5];
    for (int i = 0; i < 5; ++i) {
        Ws[i] = (const float*)d_in[4 + 2 * i];
        bs[i] = (const float*)d_in[5 + 2 * i];
    }
    float* out  = (float*)d_out;
    float* buf0 = (float*)d_ws;
    float* buf1 = buf0 + (size_t)N * 64;
    float* buf2 = buf1 + (size_t)N * 64;
    float* wpad = buf2 + (size_t)N * 64;     // 64*16 padded W4
    float* bpad = wpad + 64 * 16;            // 16 padded b4

    const dim3 ggrid((N + 15) / 16);
    const int  agrid = (E * 32 + 255) / 256;

    // Pad last-layer weights once (independent of everything until layer 4)
    gcn_pad_w4<<<(64 * 16 + 16 + 255) / 256, 256, 0, stream>>>(Ws[4], bs[4], wpad, bpad);

    // Layer 0: 128 -> 64 (no relu on input)
    gcn_gemm_wmma<<<ggrid, 128, 0, stream>>>(x, Ws[0], bs[0], buf0, N, 128, 64, 64, 0);
    gcn_zero<<<(N * 64 + 255) / 256, 256, 0, stream>>>(buf1, N * 64);
    gcn_aggregate<<<agrid, 256, 0, stream>>>(buf0, esrc, edst, ew, buf1, E, 64);

    // Layer 1: 64 -> 64 (relu of previous output fused into A load)
    gcn_gemm_wmma<<<ggrid, 128, 0, stream>>>(buf1, Ws[1], bs[1], buf2, N, 64, 64, 64, 1);
    gcn_zero<<<(N * 64 + 255) / 256, 256, 0, stream>>>(buf0, N * 64);
    gcn_aggregate<<<agrid, 256, 0, stream>>>(buf2, esrc, edst, ew, buf0, E, 64);

    // Layer 2: 64 -> 64
    gcn_gemm_wmma<<<ggrid, 128, 0, stream>>>(buf0, Ws[2], bs[2], buf1, N, 64, 64, 64, 1);
    gcn_zero<<<(N * 64 + 255) / 256, 256, 0, stream>>>(buf2, N * 64);
    gcn_aggregate<<<agrid, 256, 0, stream>>>(buf1, esrc, edst, ew, buf2, E, 64);

    // Layer 3: 64 -> 64
    gcn_gemm_wmma<<<ggrid, 128, 0, stream>>>(buf2, Ws[3], bs[3], buf0, N, 64, 64, 64, 1);
    gcn_zero<<<(N * 64 + 255) / 256, 256, 0, stream>>>(buf1, N * 64);
    gcn_aggregate<<<agrid, 256, 0, stream>>>(buf0, esrc, edst, ew, buf1, E, 64);

    // Layer 4: 64 -> 10, padded weights (doutp=16); aggregate straight to d_out
    gcn_gemm_wmma<<<ggrid, 128, 0, stream>>>(buf1, wpad, bpad, buf2, N, 64, 10, 16, 1);
    gcn_zero<<<(N * 10 + 255) / 256, 256, 0, stream>>>(out, N * 10);
    gcn_aggregate<<<agrid, 256, 0, stream>>>(buf2, esrc, edst, ew, out, E, 10);
}